// OnlineEMFitter_2018634629757
// MI455X (gfx1250) — compile-verified
//
#include <hip/hip_runtime.h>
#include <stdint.h>

// Online EM for diagonal GMM: strictly sequential scan, latency-optimized.
// 1 workgroup, 8 wave32s (wave = component k), lane = dimension d.
// CDNA5 path: async global->LDS prefetch pipeline for the permuted x-stream.

#define TAU0F   10.0f
#define EPOCHS  2
#define NPTS    200000
#define DIM     32
#define KCOMP   8
#define TSTEPS  (EPOCHS * NPTS)
#define VAR_EPS 1e-8f
#define PREF    3   // prefetch depth (s_wait_asynccnt immediates below assume 3)
#define SLOTS   4   // x ring slots; SLOTS > PREF so writes never hit the slot being read

__global__ __launch_bounds__(256, 1)
void onlineEM_scan_gfx1250(const float* __restrict__ X,
                           const float* __restrict__ pi0,
                           const float* __restrict__ mu0,
                           const float* __restrict__ var0,
                           const int*   __restrict__ perms,
                           float*       __restrict__ out)
{
    __shared__ float xbuf[SLOTS][DIM];   // prefetched x rows (written by async DMA)
    __shared__ float qsh[2][KCOMP];      // per-step quadratic forms, double-buffered
    __shared__ float pish[2][KCOMP];     // per-step pi (=s0), double-buffered

    const int tid  = threadIdx.x;
    const int k    = tid >> 5;   // wave id == component
    const int lane = tid & 31;   // lane == dimension

    // ---- init carry (consistent with reference sufficient statistics) ----
    const float pik0 = pi0[k];
    const float m0   = mu0[k * DIM + lane];
    const float v0   = var0[k * DIM + lane];
    float s0 = pik0;
    float s1 = pik0 * m0;
    float s2 = pik0 * (v0 + m0 * m0);
    float mu = m0;              // == s1/s0 at t=0
    float rv = 1.0f / v0;       // carried 1/var (var is det-1 normalized)

    const uint32_t lds_x0 = (uint32_t)(uintptr_t)(&xbuf[0][0]); // low 32 bits = LDS offset

    // ---- prime the async x pipeline: wave 0, one row (32 lanes x 4B) per step ----
    if (k == 0) {
        #pragma unroll
        for (int p = 0; p < PREF; ++p) {
            const int row   = perms[p];
            const uint64_t ga = (uint64_t)(uintptr_t)(X + (size_t)row * DIM + lane);
            const uint32_t la = lds_x0 + (uint32_t)((p & (SLOTS - 1)) * DIM + lane) * 4u;
            asm volatile("global_load_async_to_lds_b32 %0, %1, off"
                         :: "v"(la), "v"(ga) : "memory");
        }
        // PREF-1 may remain outstanding -> x[0] has landed in LDS
        asm volatile("s_wait_asynccnt 2" ::: "memory");
    }
    __syncthreads();

    for (int t = 0; t < TSTEPS; ++t) {
        const int slot = t & (SLOTS - 1);
        const int par  = t & 1;

        // ---- phase 1: E-step quadratic q_k = sum_d (x-mu)^2 / var  (log-var term == 0
        //      exactly, because var is geometric-mean-1 normalized) ----
        const float x  = xbuf[slot][lane];
        const float df = x - mu;
        float q = df * df * rv;
        #pragma unroll
        for (int m = 16; m >= 1; m >>= 1)
            q += __shfl_xor(q, m, 32);
        if (lane == 0) {
            qsh[par][k]  = q;
            pish[par][k] = s0;   // pi before this step's update
        }

        // wave 0: keep the prefetch pipeline full, off the critical path.
        // Issue x[t+PREF], then guarantee x[t+1] resident before signaling barrier.
        if (k == 0) {
            const int tp  = t + PREF;
            const int row = perms[tp < TSTEPS ? tp : (TSTEPS - 1)];
            const uint64_t ga = (uint64_t)(uintptr_t)(X + (size_t)row * DIM + lane);
            const uint32_t la = lds_x0 + (uint32_t)((tp & (SLOTS - 1)) * DIM + lane) * 4u;
            asm volatile("global_load_async_to_lds_b32 %0, %1, off"
                         :: "v"(la), "v"(ga) : "memory");
            asm volatile("s_wait_asynccnt 2" ::: "memory");
        }

        __syncthreads();   // single split barrier per step (q is double-buffered)

        // ---- phase 2: softmax over k, computed redundantly per thread from LDS ----
        float qv[KCOMP], pv[KCOMP];
        #pragma unroll
        for (int j = 0; j < KCOMP; ++j) { qv[j] = qsh[par][j]; pv[j] = pish[par][j]; }
        float qmin = qv[0];
        #pragma unroll
        for (int j = 1; j < KCOMP; ++j) qmin = fminf(qmin, qv[j]);
        float Z = 0.0f, wk = 0.0f;
        #pragma unroll
        for (int j = 0; j < KCOMP; ++j) {
            const float w = pv[j] * __expf(0.5f * (qmin - qv[j]));
            Z += w;
            if (j == k) wk = w;
        }
        const float r = wk / Z;

        // ---- Robbins-Monro update + M-step + det-1 variance regularization ----
        const float g = 1.0f / (TAU0F + (float)t);
        s0 += g * (r - s0);
        s1 += g * (r * x     - s1);
        s2 += g * (r * x * x - s2);
        const float rs0 = 1.0f / s0;
        mu = s1 * rs0;
        const float vr = fmaxf(s2 * rs0 - mu * mu, VAR_EPS);
        float lv = __logf(vr);
        #pragma unroll
        for (int m = 16; m >= 1; m >>= 1)
            lv += __shfl_xor(lv, m, 32);
        const float gm = __expf(lv * (1.0f / 32.0f)); // geomean of raw var
        rv = gm / vr;                                  // 1/var_n ; sum_d log var_n == 0
    }

    if (k == 0) asm volatile("s_wait_asynccnt 0" ::: "memory"); // drain DMA

    // ---- output: concat [pi (8), mu (8x32), var (8x32)] = 520 floats ----
    if (lane == 0) out[k] = s0;
    out[KCOMP + k * DIM + lane]               = mu;
    out[KCOMP + KCOMP * DIM + k * DIM + lane] = 1.0f / rv;
}

extern "C" void kernel_launch(void* const* d_in, const int* in_sizes, int n_in,
                              void* d_out, int out_size, void* d_ws, size_t ws_size,
                              hipStream_t stream)
{
    const float* X     = (const float*)d_in[0];
    const float* pi0   = (const float*)d_in[1];
    const float* mu0   = (const float*)d_in[2];
    const float* var0  = (const float*)d_in[3];
    const int*   perms = (const int*)  d_in[4];
    (void)in_sizes; (void)n_in; (void)out_size; (void)d_ws; (void)ws_size;

    onlineEM_scan_gfx1250<<<dim3(1), dim3(256), 0, stream>>>(
        X, pi0, mu0, var0, perms, (float*)d_out);
}